// VolumeRenderer_23862838296869
// MI455X (gfx1250) — compile-verified
//
#include <hip/hip_runtime.h>
#include <stdint.h>

// ---------------- configuration ----------------
#define ZS      128
#define CHUNK   16
#define NCHUNK  (ZS / CHUNK)
#define BLOCK   256
#define NEARF   2.0f
#define DELTAF  (4.0f / 127.0f)
#define SIG_ROW (CHUNK + 1)       // 17 dwords/row in LDS (gcd(17,64)=1 -> conflict free)
#define RAD_ROW (CHUNK * 3 + 3)   // 51 dwords/row in LDS (gcd(51,64)=1 -> conflict free)

// Select the TDM path only in the device pass and only if the builtin exists.
#if defined(__HIP_DEVICE_COMPILE__)
#if __has_builtin(__builtin_amdgcn_tensor_load_to_lds) && __has_builtin(__builtin_amdgcn_s_wait_tensorcnt)
#define DEV_TDM 1
#else
#define DEV_TDM 0
#endif
#else
#define DEV_TDM 0
#endif

#if DEV_TDM
typedef unsigned int u32x4 __attribute__((ext_vector_type(4)));
typedef int          i32x4 __attribute__((ext_vector_type(4)));
typedef int          i32x8 __attribute__((ext_vector_type(8)));

// Build a 2D tensor-DMA descriptor (D# groups 0/1 per CDNA5 ISA ch.8) and issue
// TENSOR_LOAD_TO_LDS. f32 elements, LDS padding: +1 dword after every 16 dwords.
// amdgpu-toolchain (clang-23) builtin form: 6 args
//   (u32x4 group0, i32x8 group1, i32x4, i32x4, i32x8, i32 cpol)
__device__ __forceinline__ void tdm_load_2d(uint32_t lds_off, uint64_t gaddr,
                                            uint32_t tensor_d0, uint32_t stride0,
                                            uint32_t tile_d0, uint32_t tile_d1,
                                            uint32_t tensor_d1) {
  u32x4 g0;
  g0.x = 1u;                                                  // count=1 (valid), user mode, gather off
  g0.y = lds_off;                                             // LDS byte address
  g0.z = (uint32_t)(gaddr & 0xFFFFFFFFull);                   // global_addr[31:0]
  g0.w = (uint32_t)((gaddr >> 32) & 0x01FFFFFFull)            // global_addr[56:32]
       | (2u << 30);                                          // type = 2 ("image")
  i32x8 g1;
  g1[0] = (int)((2u << 16)      // data_size = 2 (4 bytes)
              | (1u << 20)      // pad_enable
              | (3u << 22));    // pad_interval = 3 (16 dwords); pad_amount = 0 (1 dword)
  g1[1] = (int)((tensor_d0 & 0xFFFFu) << 16);                               // tensor_dim0[15:0]
  g1[2] = (int)(((tensor_d0 >> 16) & 0xFFFFu) | ((tensor_d1 & 0xFFFFu) << 16));
  g1[3] = (int)(((tensor_d1 >> 16) & 0xFFFFu) | ((tile_d0 & 0xFFFFu) << 16));
  g1[4] = (int)(tile_d1 & 0xFFFFu);                                         // tile_dim1, tile_dim2=0
  g1[5] = (int)stride0;                                                     // tensor_dim0_stride[31:0]
  g1[6] = 0;                                                                // stride0 hi / stride1 lo
  g1[7] = 0;
  const i32x4 z4 = {0, 0, 0, 0};                 // groups 2/3: tile_dim3/4 = 0 -> unused
  const i32x8 z8 = {0, 0, 0, 0, 0, 0, 0, 0};     // extra group (zero-filled, probe-verified)
  __builtin_amdgcn_tensor_load_to_lds(g0, g1, z4, z4, z8, 0);
}
#endif  // DEV_TDM

__global__ __launch_bounds__(BLOCK) void VolumeRenderer_kernel(
    const float* __restrict__ sigmas, const float* __restrict__ radiances,
    float* __restrict__ rgb_out, float* __restrict__ depth_out, int nrays) {
#if DEV_TDM
  __shared__ float s_sig[2][BLOCK * SIG_ROW];   //  34816 B
  __shared__ float s_rad[2][BLOCK * RAD_ROW];   // 104448 B  (total 136 KB -> 2 blocks/WGP)

  const int tid    = threadIdx.x;
  const int r0     = blockIdx.x * BLOCK;
  const bool issuer = (tid < 32);               // wave 0 issues DMA (wave-uniform branch)

  const uint64_t sig_ga0 = (uint64_t)(uintptr_t)sigmas    + (uint64_t)r0 * (ZS * 4u);
  const uint64_t rad_ga0 = (uint64_t)(uintptr_t)radiances + (uint64_t)r0 * (ZS * 12u);

  if (issuer) {
#pragma unroll
    for (int p = 0; p < 2; ++p) {
      tdm_load_2d((uint32_t)(uintptr_t)&s_sig[p][0],
                  sig_ga0 + (uint64_t)p * (CHUNK * 4u),
                  ZS, ZS, CHUNK, BLOCK, (uint32_t)nrays);
      tdm_load_2d((uint32_t)(uintptr_t)&s_rad[p][0],
                  rad_ga0 + (uint64_t)p * (CHUNK * 12u),
                  ZS * 3, ZS * 3, CHUNK * 3, BLOCK, (uint32_t)nrays);
    }
  }

  float T = 1.0f, cr = 0.0f, cg = 0.0f, cb = 0.0f, cd = 0.0f;

#pragma unroll
  for (int k = 0; k < NCHUNK; ++k) {
    const int p = k & 1;
    if (issuer) {
      if (k < NCHUNK - 1) __builtin_amdgcn_s_wait_tensorcnt(2);  // pair k complete
      else                __builtin_amdgcn_s_wait_tensorcnt(0);
    }
    __syncthreads();   // buffer p ready for all waves

    const float* sg = &s_sig[p][tid * SIG_ROW];
    const float* rd = &s_rad[p][tid * RAD_ROW];
#pragma unroll
    for (int j = 0; j < CHUNK; ++j) {
      const int   z    = k * CHUNK + j;
      const float dist = (z == ZS - 1) ? 1e10f : DELTAF;
      const float e    = __expf(-sg[j] * dist);   // exp(-sigma*dist)
      T *= (e + 1e-10f);                          // inclusive cumprod (matches reference)
      const float w = (1.0f - e) * T;
      const int p0 = 3 * j, p1 = 3 * j + 1, p2 = 3 * j + 2;   // +1 pad dword per 16 stored
      cr = fmaf(w, rd[p0 + (p0 >> 4)], cr);
      cg = fmaf(w, rd[p1 + (p1 >> 4)], cg);
      cb = fmaf(w, rd[p2 + (p2 >> 4)], cb);
      cd = fmaf(w, NEARF + (float)z * DELTAF, cd);
    }

    __syncthreads();   // all waves done reading buffer p
    if (issuer && (k + 2) < NCHUNK) {
      const uint64_t cs = (uint64_t)(k + 2) * (CHUNK * 4u);
      tdm_load_2d((uint32_t)(uintptr_t)&s_sig[p][0], sig_ga0 + cs,
                  ZS, ZS, CHUNK, BLOCK, (uint32_t)nrays);
      tdm_load_2d((uint32_t)(uintptr_t)&s_rad[p][0], rad_ga0 + cs * 3u,
                  ZS * 3, ZS * 3, CHUNK * 3, BLOCK, (uint32_t)nrays);
    }
  }

  const int g = r0 + tid;
  rgb_out[3 * g + 0] = cr;
  rgb_out[3 * g + 1] = cg;
  rgb_out[3 * g + 2] = cb;
  depth_out[g]       = cd;
#else
  // Fallback: direct streaming, one thread per ray, float4 loads.
  const int g = blockIdx.x * BLOCK + threadIdx.x;
  const float4* sg4 = (const float4*)(sigmas    + (size_t)g * ZS);
  const float4* rd4 = (const float4*)(radiances + (size_t)g * ZS * 3);
  float T = 1.0f, cr = 0.0f, cg = 0.0f, cb = 0.0f, cd = 0.0f;
#pragma unroll 4
  for (int q = 0; q < ZS / 4; ++q) {
    const float4 s  = sg4[q];
    const float4 ra = rd4[3 * q + 0];
    const float4 rb = rd4[3 * q + 1];
    const float4 rc = rd4[3 * q + 2];
    const float sv[4]  = {s.x, s.y, s.z, s.w};
    const float rv[12] = {ra.x, ra.y, ra.z, ra.w, rb.x, rb.y, rb.z, rb.w, rc.x, rc.y, rc.z, rc.w};
#pragma unroll
    for (int u = 0; u < 4; ++u) {
      const int   z    = 4 * q + u;
      const float dist = (z == ZS - 1) ? 1e10f : DELTAF;
      const float e    = __expf(-sv[u] * dist);
      T *= (e + 1e-10f);
      const float w = (1.0f - e) * T;
      cr = fmaf(w, rv[3 * u + 0], cr);
      cg = fmaf(w, rv[3 * u + 1], cg);
      cb = fmaf(w, rv[3 * u + 2], cb);
      cd = fmaf(w, NEARF + (float)z * DELTAF, cd);
    }
  }
  rgb_out[3 * g + 0] = cr;
  rgb_out[3 * g + 1] = cg;
  rgb_out[3 * g + 2] = cb;
  depth_out[g]       = cd;
#endif
}

extern "C" void kernel_launch(void* const* d_in, const int* in_sizes, int n_in,
                              void* d_out, int out_size, void* d_ws, size_t ws_size,
                              hipStream_t stream) {
  const float* sigmas    = (const float*)d_in[0];   // [B,R,Z,1] f32
  const float* radiances = (const float*)d_in[1];   // [B,R,Z,3] f32
  float* out = (float*)d_out;

  const int nrays = in_sizes[0] / ZS;               // B*R = 262144
  float* rgb   = out;                               // [B,R,3] flat
  float* depth = out + (size_t)nrays * 3;           // [B,R,1] flat

  dim3 grid(nrays / BLOCK), block(BLOCK);
  VolumeRenderer_kernel<<<grid, block, 0, stream>>>(sigmas, radiances, rgb, depth, nrays);
}